// CRFLayer_84877143703938
// MI455X (gfx1250) — compile-verified
//
#include <hip/hip_runtime.h>
#include <cstdint>
#include <cstddef>

#define Bn 256
#define Sn 1024
#define Tn 64
#define PSTR 68                    // padded LDS row stride: conflict-free b64 A reads
#define INV_LN2f 1.44269504088896341f
#define LN2f     0.69314718055994531f
#define LOG2_64f 6.0f              // log2(64), exact

typedef float v2f __attribute__((ext_vector_type(2)));
typedef float v4f __attribute__((ext_vector_type(4)));
typedef float v8f __attribute__((ext_vector_type(8)));

// ---------------------------------------------------------------- E = exp(T)
__global__ __launch_bounds__(256) void crf_prep(const float* __restrict__ trans,
                                                float* __restrict__ E) {
    int i = blockIdx.x * 256 + threadIdx.x;
    if (i < Tn * Tn) E[i] = __expf(trans[i]);
}

// ---------------------------------------------------------------- forward scan
// Block = 16 batches, 4 waves; wave w owns states [16w,16w+16).
// Base-2 log domain: p = 2^z2 in ping-pong LDS; per-step update is
//   p' = s * 2^(e/ln2 - zc - 6)   with s = P x E via f32 WMMA,
// so the only post-WMMA op on the critical path is one multiply.
__global__ __launch_bounds__(128) void crf_scan(const float* __restrict__ emis,
                                                const uint8_t* __restrict__ mask,
                                                const float* __restrict__ E,
                                                float* __restrict__ fwd) {
    __shared__ __align__(16) float   pbuf[2][16 * PSTR];
    __shared__ __align__(16) float   zc[2][16];
    __shared__ __align__(8)  uint8_t mT[Sn][16];   // mask transposed: [t][batch]

    const int tid    = threadIdx.x;
    const int wave   = tid >> 5;
    const int lane   = tid & 31;
    const int lane16 = lane & 15;
    const int half   = lane >> 4;
    const int n0     = wave * 16;
    const int bt     = blockIdx.x * 16;
    const int col    = n0 + lane16;
    const int br0    = 8 * half + (lane16 & 7);    // batch tracked by this thread's Mreg

    // B fragments (E column tile) resident in 32 VGPRs for the whole scan.
    v2f bb[16];
#pragma unroll
    for (int c = 0; c < 16; ++c) {
        int kb  = 4 * c + 2 * half;
        bb[c].x = E[kb * Tn + col];
        bb[c].y = E[(kb + 1) * Tn + col];
    }

    // Stage mask (transposed) to LDS; init z2 = alpha0/ln2, p = 2^z2.
    for (int idx = tid; idx < 16 * Sn; idx += 128) {
        int b = idx >> 10, s = idx & (Sn - 1);
        mT[s][b] = mask[(size_t)(bt + b) * Sn + s];
    }
    for (int idx = tid; idx < 16 * Tn; idx += 128) {
        int b = idx >> 6, j = idx & 63;
        float z = emis[((size_t)(bt + b) * Sn) * Tn + j] * INV_LN2f;
        pbuf[0][b * PSTR + j] = __builtin_amdgcn_exp2f(z);
        if (j == 0) zc[0][b] = z;
    }
    float Mreg = 0.f;
    __syncthreads();

    // Per-row emission pointers (row r -> batch r + 8*half), at t=1.
    const float* eptr[8];
#pragma unroll
    for (int r = 0; r < 8; ++r)
        eptr[r] = emis + ((size_t)(bt + r + 8 * half) * Sn + 1) * Tn + col;

    float eA[8], eB[8];
#pragma unroll
    for (int r = 0; r < 8; ++r) eA[r] = eptr[r][0];

    auto step = [&](int t, const float* __restrict__ psrc, float* __restrict__ pdst,
                    const float* __restrict__ zsrc, float* __restrict__ zdst,
                    float (&eu)[8], float (&el)[8]) {
        // 8 mask bytes for this half's batches: single b64 LDS load
        unsigned long long mv = *(const unsigned long long*)&mT[t][8 * half];

        // z2 at state 0 for this half's 8 batches: two b128 LDS loads
        v4f c0 = *(const v4f*)&zsrc[8 * half];
        v4f c1 = *(const v4f*)&zsrc[8 * half + 4];
        float czc[8] = {c0.x, c0.y, c0.z, c0.w, c1.x, c1.y, c1.z, c1.w};

        // offset accumulation (redundant per thread; no exec churn)
        {
            float Mz = zsrc[br0] + LOG2_64f;
            Mreg += ((mv >> (8 * (lane16 & 7))) & 1) ? Mz : 0.f;
        }

        // emission scale factors: independent of s -> TRANS overlaps the WMMAs
        float d[8], f[8];
#pragma unroll
        for (int r = 0; r < 8; ++r) {
            d[r] = eu[r] * INV_LN2f - (czc[r] + LOG2_64f);
            f[r] = __builtin_amdgcn_exp2f(d[r]);
        }

        // s = P x E : 4 independent WMMA chains of depth 4
        v8f a0 = {0,0,0,0,0,0,0,0}, a1 = a0, a2 = a0, a3 = a0;
        const float* prow = &psrc[lane16 * PSTR + 2 * half];
#pragma unroll
        for (int c = 0; c < 4; ++c) {
            v2f x0 = *(const v2f*)(prow + 4 * c);
            v2f x1 = *(const v2f*)(prow + 4 * (c + 4));
            v2f x2 = *(const v2f*)(prow + 4 * (c + 8));
            v2f x3 = *(const v2f*)(prow + 4 * (c + 12));
            a0 = __builtin_amdgcn_wmma_f32_16x16x4_f32(false, x0, false, bb[c],      (short)0, a0, false, false);
            a1 = __builtin_amdgcn_wmma_f32_16x16x4_f32(false, x1, false, bb[c + 4],  (short)0, a1, false, false);
            a2 = __builtin_amdgcn_wmma_f32_16x16x4_f32(false, x2, false, bb[c + 8],  (short)0, a2, false, false);
            a3 = __builtin_amdgcn_wmma_f32_16x16x4_f32(false, x3, false, bb[c + 12], (short)0, a3, false, false);
        }
        v8f acc = (a0 + a1) + (a2 + a3);

        // p' = s * 2^d  (single mul per row on the critical path); masked: copy old p
#pragma unroll
        for (int r = 0; r < 8; ++r) {
            int   br = r + 8 * half;
            float pn = ((mv >> (8 * r)) & 1) ? acc[r] * f[r] : psrc[br * PSTR + col];
            pdst[br * PSTR + col] = pn;
        }

        // state-0 recentering value: log2 only on the 2 lanes that own column 0
        if (wave == 0 && lane16 == 0) {
#pragma unroll
            for (int r = 0; r < 8; ++r)
                zdst[r + 8 * half] = ((mv >> (8 * r)) & 1)
                                   ? __builtin_amdgcn_logf(acc[r]) + d[r]
                                   : czc[r];
        }

        // next-step emissions + prefetch: issue before the barrier so the
        // loads progress while waiting (syncthreads only drains dscnt)
        const int tn = (t + 1 < Sn) ? (t + 1) : (Sn - 1);
#pragma unroll
        for (int r = 0; r < 8; ++r) {
            el[r] = eptr[r][(size_t)(tn - 1) * Tn];
            if (t + 8 < Sn) __builtin_prefetch(eptr[r] + (size_t)(t + 7) * Tn, 0, 1);
        }
        __syncthreads();
    };

    // 1023 steps, unrolled x2 so ping-pong addresses are compile-time constant
    for (int tb = 1; tb < Sn; tb += 2) {
        step(tb, pbuf[0], pbuf[1], zc[0], zc[1], eA, eB);
        if (tb + 1 < Sn)
            step(tb + 1, pbuf[1], pbuf[0], zc[1], zc[0], eB, eA);
    }

    // fwd[b] = ln2 * (M2_b + log2(sum_j p[b,j]));  final p lives in pbuf[1]
    if (wave == 0 && lane16 < 8) {
        int b = 8 * half + lane16;
        v4f s4 = {0, 0, 0, 0};
#pragma unroll
        for (int j = 0; j < 16; ++j) s4 += *(const v4f*)&pbuf[1][b * PSTR + 4 * j];
        float s = (s4.x + s4.y) + (s4.z + s4.w);
        fwd[bt + b] = LN2f * (Mreg + __builtin_amdgcn_logf(s));
    }
}

// ---------------------------------------------------------------- gold score
__global__ __launch_bounds__(128) void crf_gold(const float* __restrict__ emis,
                                                const int* __restrict__ tags,
                                                const uint8_t* __restrict__ mask,
                                                const float* __restrict__ trans,
                                                float* __restrict__ gold) {
    const int b = blockIdx.x, tid = threadIdx.x;
    float acc = 0.f;
    for (int s = tid; s < Sn; s += 128) {
        int   tg = tags[(size_t)b * Sn + s];
        float m  = mask[(size_t)b * Sn + s] ? 1.f : 0.f;
        acc += m * emis[((size_t)b * Sn + s) * Tn + tg];
        if (s >= 1) {
            int tp = tags[(size_t)b * Sn + s - 1];
            acc += m * trans[tp * Tn + tg];
        }
    }
    __shared__ float red[128];
    red[tid] = acc;
    __syncthreads();
    for (int off = 64; off > 0; off >>= 1) {
        if (tid < off) red[tid] += red[tid + off];
        __syncthreads();
    }
    if (tid == 0) gold[b] = red[0];
}

// ---------------------------------------------------------------- mean(fwd-gold)
__global__ __launch_bounds__(256) void crf_final(const float* __restrict__ fwd,
                                                 const float* __restrict__ gold,
                                                 float* __restrict__ out) {
    __shared__ float red[256];
    int tid = threadIdx.x;
    red[tid] = fwd[tid] - gold[tid];
    __syncthreads();
    for (int off = 128; off > 0; off >>= 1) {
        if (tid < off) red[tid] += red[tid + off];
        __syncthreads();
    }
    if (tid == 0) out[0] = red[0] * (1.f / Bn);
}

extern "C" void kernel_launch(void* const* d_in, const int* in_sizes, int n_in,
                              void* d_out, int out_size, void* d_ws, size_t ws_size,
                              hipStream_t stream) {
    const float*   emis  = (const float*)d_in[0];
    const int*     tags  = (const int*)d_in[1];
    const uint8_t* mask  = (const uint8_t*)d_in[2];
    const float*   trans = (const float*)d_in[3];

    float* E    = (float*)d_ws;          // 64*64 f32
    float* fwd  = E + Tn * Tn;           // 256 f32
    float* gold = fwd + Bn;              // 256 f32

    crf_prep <<<16,      256, 0, stream>>>(trans, E);
    crf_scan <<<Bn / 16, 128, 0, stream>>>(emis, mask, E, fwd);
    crf_gold <<<Bn,      128, 0, stream>>>(emis, tags, mask, trans, gold);
    crf_final<<<1,       256, 0, stream>>>(fwd, gold, (float*)d_out);
}